// Linear_act_sp_35622458753937
// MI455X (gfx1250) — compile-verified
//
#include <hip/hip_runtime.h>
#include <stdint.h>

// ---------------------------------------------------------------------------
// 2:4 activation-sparse linear:  out[M,N] = prune_2of4(x)[M,K] @ W[N,K]^T
// M=8192, K=4096, N=4096.
// bf16 V_SWMMAC (2:4 sparse) + async global->LDS double-buffered staging.
// ---------------------------------------------------------------------------

#define M_DIM 8192
#define K_DIM 4096
#define N_DIM 4096
#define KC (K_DIM / 2)    // compressed K (bf16 elements per row of A_c)
#define KGU (K_DIM / 32)  // packed-index u32 words per row (8 groups / u32)

typedef __attribute__((ext_vector_type(4)))  float    v4f;
typedef __attribute__((ext_vector_type(4)))  uint32_t v4u;
typedef __attribute__((ext_vector_type(4)))  int      v4i;
typedef __attribute__((ext_vector_type(8)))  float    v8f;
typedef __attribute__((ext_vector_type(16))) __bf16   v16bf;
typedef __attribute__((ext_vector_type(32))) __bf16   v32bf;

union FragA { v4u q[2]; v16bf v; };   // 16x32 bf16 A (compressed), 8 VGPRs
union FragB { v4u q[4]; v32bf v; };   // 64x16 bf16 B (dense), 16 VGPRs

// ---- CDNA5 async global->LDS path (probed; safe fallbacks keep build green)
#if __has_builtin(__builtin_amdgcn_global_load_async_to_lds_b128)
#define HAS_ASYNC 1
#endif

typedef __attribute__((address_space(1))) v4i GV4;   // global int4*
typedef __attribute__((address_space(3))) v4i LV4;   // LDS int4*

__device__ __forceinline__ void cp16_async(const void* gsrc, void* ldst) {
#ifdef HAS_ASYNC
  __builtin_amdgcn_global_load_async_to_lds_b128((GV4*)gsrc, (LV4*)ldst, 0, 0);
#else
  *(v4u*)ldst = *(const v4u*)gsrc;   // vgpr load + ds_store fallback
#endif
}

#ifdef HAS_ASYNC
#if __has_builtin(__builtin_amdgcn_s_wait_asynccnt)
#define WAIT_ASYNC(n) __builtin_amdgcn_s_wait_asynccnt(n)
#else
#define WAIT_ASYNC(n) asm volatile("s_wait_asynccnt %0" ::"i"(n))
#endif
#else
#define WAIT_ASYNC(n) ((void)0)
#endif

__device__ __forceinline__ uint32_t f2bf_rne(float f) {
  uint32_t u = __builtin_bit_cast(uint32_t, f);
  return (u + 0x7FFFu + ((u >> 16) & 1u)) >> 16;   // round-to-nearest-even
}

// ---------------------------------------------------------------------------
// Stage 0: W fp32 -> bf16 (row-major [N,K]).  8 elems / thread.
// ---------------------------------------------------------------------------
__global__ __launch_bounds__(256) void k_convert_w(const float* __restrict__ w,
                                                   uint16_t* __restrict__ wbf) {
  size_t t = (size_t)blockIdx.x * 256 + threadIdx.x;     // N*K/8 threads
  const v4f* p = (const v4f*)(w + t * 8);
  v4f a = p[0], b = p[1];
  v4u o;
  o.x = f2bf_rne(a.x) | (f2bf_rne(a.y) << 16);
  o.y = f2bf_rne(a.z) | (f2bf_rne(a.w) << 16);
  o.z = f2bf_rne(b.x) | (f2bf_rne(b.y) << 16);
  o.w = f2bf_rne(b.z) | (f2bf_rne(b.w) << 16);
  *(v4u*)(wbf + t * 8) = o;
}

// ---------------------------------------------------------------------------
// Stage 1: prune 2:4 + compress (stable-argsort tie-breaking).
//   x fp32 [M,K] -> A_c bf16 [M,KC] + idx u32 [M,KGU] (8x 4-bit codes/u32,
//   nibble order matches SWMMAC sparse-index VGPR layout).
// ---------------------------------------------------------------------------
__global__ __launch_bounds__(256) void k_prune_compress(
    const float* __restrict__ x, uint16_t* __restrict__ ac,
    uint32_t* __restrict__ idxw) {
  size_t t = (size_t)blockIdx.x * 256 + threadIdx.x;     // M*K/32 threads
  int row = (int)(t >> 7);                                // 128 chunks per row
  int ch  = (int)(t & 127);
  const v4f* px = (const v4f*)(x + (size_t)row * K_DIM + (size_t)ch * 32);
  uint32_t packed[8];
  uint32_t codes = 0;
#pragma unroll
  for (int g = 0; g < 8; ++g) {
    v4f v = px[g];
    float val[4] = {v.x, v.y, v.z, v.w};
    float av[4]  = {fabsf(v.x), fabsf(v.y), fabsf(v.z), fabsf(v.w)};
    int keep0 = -1, keep1 = -1;
#pragma unroll
    for (int i = 0; i < 4; ++i) {
      int r = 0;
#pragma unroll
      for (int j = 0; j < 4; ++j)
        r += (av[j] < av[i]) || ((av[j] == av[i]) && (j < i));
      if (r >= 2) { if (keep0 < 0) keep0 = i; else keep1 = i; }
    }
    packed[g] = f2bf_rne(val[keep0]) | (f2bf_rne(val[keep1]) << 16);
    codes |= (uint32_t)(keep0 | (keep1 << 2)) << (g * 4);
  }
  v4u* pac = (v4u*)(ac + (size_t)row * KC + (size_t)ch * 16);
  v4u q0 = {packed[0], packed[1], packed[2], packed[3]};
  v4u q1 = {packed[4], packed[5], packed[6], packed[7]};
  pac[0] = q0;
  pac[1] = q1;
  idxw[(size_t)row * KGU + ch] = codes;
}

// ---------------------------------------------------------------------------
// Stage 2: sparse GEMM.  Block = 256 thr = 8 waves, C tile 128x128.
// Wave grid 2(M) x 4(N); each wave: 4 M-tiles x 2 N-tiles of 16x16 C frags.
// Per K-step (64 expanded K): async-copy A(8KB)+B(16KB) tile into LDS
// (double-buffered, padded strides), ds_load fragments, 8 swmmac / wave.
// ---------------------------------------------------------------------------
#define A_STRIDE 72            // 64B row + 8B pad (bank-conflict-free)
#define B_STRIDE 136           // 128B row + 8B pad
#define A_BYTES (128 * A_STRIDE)
#define B_BYTES (128 * B_STRIDE)
#define BUF_BYTES (A_BYTES + B_BYTES)
#define CP_PER_THREAD 6        // 2 A-chunks + 4 B-chunks of 16B

__global__ __launch_bounds__(256, 1) void k_gemm_sp(
    const uint16_t* __restrict__ ac, const uint32_t* __restrict__ idxw,
    const uint16_t* __restrict__ wbf, float* __restrict__ out) {
  __shared__ __align__(16) char lds[2][BUF_BYTES];

  const int tid  = threadIdx.x;
  const int lane = tid & 31;
  const int wave = tid >> 5;
  const int wm = wave & 1;        // 2 waves along M
  const int wn = wave >> 1;       // 4 waves along N
  const int m0 = blockIdx.y * 128 + wm * 64;
  const int n0 = blockIdx.x * 128 + wn * 32;
  const int l16 = lane & 15;      // A: row-in-tile, B/C: col(N)
  const int h   = lane >> 4;      // lane half selects K/M sub-range

  // ---- per-thread async-copy source pointers (advance by constants) ------
  // A tile: 128 rows x 64B (32 compressed bf16).  512 chunks of 16B.
  const char* gA[2];
  int lA[2];
#pragma unroll
  for (int i = 0; i < 2; ++i) {
    int c = tid + i * 256;
    int row = c >> 2, off = c & 3;
    gA[i] = (const char*)(ac + (size_t)(blockIdx.y * 128 + row) * KC + off * 8);
    lA[i] = row * A_STRIDE + off * 16;
  }
  // B tile: 128 rows x 128B (64 bf16).  1024 chunks of 16B.
  const char* gB[4];
  int lB[4];
#pragma unroll
  for (int i = 0; i < 4; ++i) {
    int c = tid + i * 256;
    int row = c >> 3, off = c & 7;
    gB[i] = (const char*)(wbf + (size_t)(blockIdx.x * 128 + row) * K_DIM + off * 8);
    lB[i] = A_BYTES + row * B_STRIDE + off * 16;
  }

  auto issue_tile = [&](int buf) {
    char* dst = &lds[buf][0];
#pragma unroll
    for (int i = 0; i < 2; ++i) { cp16_async(gA[i], dst + lA[i]); gA[i] += 64; }
#pragma unroll
    for (int i = 0; i < 4; ++i) { cp16_async(gB[i], dst + lB[i]); gB[i] += 128; }
  };

  // ---- sparse-index source pointers (small; direct global per-lane) ------
  const uint32_t* iP[4];
#pragma unroll
  for (int mt = 0; mt < 4; ++mt)
    iP[mt] = idxw + (size_t)(m0 + mt * 16 + l16) * KGU + h;

  v8f c[4][2];
#pragma unroll
  for (int mt = 0; mt < 4; ++mt)
#pragma unroll
    for (int nt = 0; nt < 2; ++nt) c[mt][nt] = (v8f)0.0f;

  issue_tile(0);                 // prologue: stage first K-step
  int cur = 0;

  for (int k0 = 0; k0 < K_DIM; k0 += 64) {
    const bool more = (k0 + 64) < K_DIM;
    if (more) {
      issue_tile(cur ^ 1);       // prefetch next K-step into other buffer
      WAIT_ASYNC(CP_PER_THREAD); // previous tile's copies complete
    } else {
      WAIT_ASYNC(0);
    }
    __syncthreads();

    const char* bufA = &lds[cur][0];
    const char* bufB = &lds[cur][A_BYTES];

    FragA a[4];
    int   ai[4];
#pragma unroll
    for (int mt = 0; mt < 4; ++mt) {
      // A 16-bit 16x32 layout: lane m -> V0..3 = cK[h*8..+7], V4..7 = cK[16+h*8..]
      const char* base = bufA + (mt * 16 + l16) * A_STRIDE + h * 16;
      a[mt].q[0] = *(const v4u*)(base);
      a[mt].q[1] = *(const v4u*)(base + 32);
      ai[mt] = (int)iP[mt][k0 >> 5];
    }
    FragB b[2];
#pragma unroll
    for (int nt = 0; nt < 2; ++nt) {
      // B 64x16 layout: lane (n,h) -> V0..7 = K[h*16..+15], V8..15 = K[32+h*16..]
      const char* base = bufB + (nt * 16 + l16) * B_STRIDE + h * 32;
      b[nt].q[0] = *(const v4u*)(base);
      b[nt].q[1] = *(const v4u*)(base + 16);
      b[nt].q[2] = *(const v4u*)(base + 64);
      b[nt].q[3] = *(const v4u*)(base + 80);
    }

#pragma unroll
    for (int mt = 0; mt < 4; ++mt)
#pragma unroll
      for (int nt = 0; nt < 2; ++nt)
        c[mt][nt] = __builtin_amdgcn_swmmac_f32_16x16x64_bf16(
            false, a[mt].v, false, b[nt].v, c[mt][nt], ai[mt], false, false);

    __syncthreads();             // all waves done reading before buffer reuse
    cur ^= 1;
  }

  // C layout: lanes 0-15 N=lane M=0..7 in V0..7; lanes 16-31 N=lane-16 M=8..15
#pragma unroll
  for (int mt = 0; mt < 4; ++mt)
#pragma unroll
    for (int nt = 0; nt < 2; ++nt) {
      float* po = out + (size_t)(m0 + mt * 16 + h * 8) * N_DIM + n0 + nt * 16 + l16;
#pragma unroll
      for (int r = 0; r < 8; ++r) po[(size_t)r * N_DIM] = c[mt][nt][r];
    }
}

// ---------------------------------------------------------------------------
extern "C" void kernel_launch(void* const* d_in, const int* in_sizes, int n_in,
                              void* d_out, int out_size, void* d_ws, size_t ws_size,
                              hipStream_t stream) {
  const float* x = (const float*)d_in[0];   // [2,4096,4096] fp32
  const float* w = (const float*)d_in[1];   // [4096,4096]  fp32
  float* out = (float*)d_out;               // [8192,4096]  fp32

  char* ws = (char*)d_ws;
  uint16_t* wbf = (uint16_t*)ws;                                   // 32 MiB
  uint16_t* acb = (uint16_t*)(ws + (size_t)N_DIM * K_DIM * 2);     // 32 MiB
  uint32_t* idx = (uint32_t*)(ws + (size_t)N_DIM * K_DIM * 2
                                 + (size_t)M_DIM * KC * 2);        //  4 MiB

  {  // W fp32 -> bf16
    size_t thr = (size_t)N_DIM * K_DIM / 8;
    k_convert_w<<<dim3((unsigned)(thr / 256)), dim3(256), 0, stream>>>(w, wbf);
  }
  {  // prune + compress activations
    size_t thr = (size_t)M_DIM * K_DIM / 32;
    k_prune_compress<<<dim3((unsigned)(thr / 256)), dim3(256), 0, stream>>>(x, acb, idx);
  }
  {  // sparse GEMM
    dim3 grid(N_DIM / 128, M_DIM / 128);
    k_gemm_sp<<<grid, dim3(256), 0, stream>>>(acb, idx, wbf, out);
  }
}